// _test_GNN_49709951484254
// MI455X (gfx1250) — compile-verified
//
#include <hip/hip_runtime.h>

#define N_NODES 50000
#define E_EDGES 800000
#define ETOT    (E_EDGES + N_NODES)   // edges + self loops = 850000
#define F_INK   128                   // K of every GEMM (F_IN and H*C are both 128)
#define HEADS   4
#define CHID    32
#define NCLS    40
#define W_HID   (HEADS * CHID)        // 128
#define W_OUT   (HEADS * NCLS)        // 160
#define NEG_SLOPE 0.2f

typedef float v2f __attribute__((ext_vector_type(2)));
typedef float v8f __attribute__((ext_vector_type(8)));

// ---------------------------------------------------------------------------
// GEMM: C[M x Nw] = A[M x 128] * B[128 x Nw] via V_WMMA_F32_16X16X4_F32.
// One wave computes one 16x16 tile. grid.x = M/16 row tiles; wave id = col tile.
// A frag (16x4, MxK): lane l in group g=(l>>4): row = l&15, VGPR r holds K=2g+r.
// B frag (4x16, KxN): lane l: col = l&15, VGPR r holds K=2g+r (row-striped).
// C/D (16x16): VGPR r: row = r + 8g, col = l&15.
// ---------------------------------------------------------------------------
__global__ void gemm_wmma_f32(const float* __restrict__ A,
                              const float* __restrict__ B,
                              float* __restrict__ C,
                              int Nw) {
  const int wave = threadIdx.x >> 5;
  const int lane = threadIdx.x & 31;
  const int g = lane >> 4;      // 0 or 1
  const int l = lane & 15;
  const int m0 = blockIdx.x * 16;
  const int n0 = wave * 16;

  const float* __restrict__ arow = A + (size_t)(m0 + l) * F_INK;
  v8f acc = {};
#pragma unroll 4
  for (int k = 0; k < F_INK; k += 4) {
    v2f a, b;
    a.x = arow[k + 2 * g + 0];
    a.y = arow[k + 2 * g + 1];
    b.x = B[(size_t)(k + 2 * g + 0) * Nw + n0 + l];
    b.y = B[(size_t)(k + 2 * g + 1) * Nw + n0 + l];
    acc = __builtin_amdgcn_wmma_f32_16x16x4_f32(
        /*neg_a=*/false, a, /*neg_b=*/false, b,
        /*c_mod=*/(short)0, acc, /*reuse_a=*/false, /*reuse_b=*/false);
  }
#pragma unroll
  for (int r = 0; r < 8; ++r) {
    C[(size_t)(m0 + r + 8 * g) * Nw + (n0 + l)] = acc[r];
  }
}

// ---------------------------------------------------------------------------
// Per-(node, head) attention scalars: es = <h, a_src>, ed = <h, a_dst>
// ---------------------------------------------------------------------------
__global__ void att_scalars(const float* __restrict__ h,
                            const float* __restrict__ asrc,
                            const float* __restrict__ adst,
                            float* __restrict__ es, float* __restrict__ ed,
                            int Ch) {
  int t = blockIdx.x * blockDim.x + threadIdx.x;
  if (t >= N_NODES * HEADS) return;
  int n = t / HEADS, hd = t % HEADS;
  const float* hp = h + (size_t)n * HEADS * Ch + (size_t)hd * Ch;
  const float* as = asrc + hd * Ch;
  const float* ad = adst + hd * Ch;
  float s = 0.f, d = 0.f;
  for (int c = 0; c < Ch; ++c) {
    float v = hp[c];
    s += v * as[c];
    d += v * ad[c];
  }
  es[t] = s;
  ed[t] = d;
}

// order-preserving float <-> uint mapping (for atomic segment-max)
__device__ __forceinline__ unsigned orderKey(float f) {
  int i = __float_as_int(f);
  return (i >= 0) ? ((unsigned)i ^ 0x80000000u) : ~(unsigned)i;
}
__device__ __forceinline__ float keyToFloat(unsigned k) {
  int i = (k & 0x80000000u) ? (int)(k ^ 0x80000000u) : (int)(~k);
  return __int_as_float(i);
}

__device__ __forceinline__ void edge_endpoints(int e, const int* __restrict__ src,
                                               const int* __restrict__ dst,
                                               int& s, int& d) {
  if (e < E_EDGES) { s = src[e]; d = dst[e]; }
  else             { s = e - E_EDGES; d = s; }   // self loop
}

// one thread per edge; all HEADS(=4) handled with float4 loads
__global__ void edge_max(const int* __restrict__ src, const int* __restrict__ dst,
                         const float* __restrict__ es, const float* __restrict__ ed,
                         unsigned* __restrict__ mkey) {
  int e = blockIdx.x * blockDim.x + threadIdx.x;
  if (e >= ETOT) return;
  int s, d;
  edge_endpoints(e, src, dst, s, d);
  float4 vs = *(const float4*)(es + (size_t)s * HEADS);
  float4 vd = *(const float4*)(ed + (size_t)d * HEADS);
  float a[4] = {vs.x + vd.x, vs.y + vd.y, vs.z + vd.z, vs.w + vd.w};
#pragma unroll
  for (int i = 0; i < HEADS; ++i) {
    float v = a[i];
    v = (v > 0.f) ? v : NEG_SLOPE * v;          // leaky relu
    atomicMax(&mkey[d * HEADS + i], orderKey(v));
  }
}

__global__ void edge_expsum(const int* __restrict__ src, const int* __restrict__ dst,
                            const float* __restrict__ es, const float* __restrict__ ed,
                            const unsigned* __restrict__ mkey,
                            float* __restrict__ ealpha, float* __restrict__ denom) {
  int e = blockIdx.x * blockDim.x + threadIdx.x;
  if (e >= ETOT) return;
  int s, d;
  edge_endpoints(e, src, dst, s, d);
  float4 vs = *(const float4*)(es + (size_t)s * HEADS);
  float4 vd = *(const float4*)(ed + (size_t)d * HEADS);
  uint4  mk = *(const uint4*)(mkey + (size_t)d * HEADS);
  float a[4] = {vs.x + vd.x, vs.y + vd.y, vs.z + vd.z, vs.w + vd.w};
  unsigned k[4] = {mk.x, mk.y, mk.z, mk.w};
  float ea[4];
#pragma unroll
  for (int i = 0; i < HEADS; ++i) {
    float v = a[i];
    v = (v > 0.f) ? v : NEG_SLOPE * v;
    ea[i] = __expf(v - keyToFloat(k[i]));
    atomicAdd(&denom[d * HEADS + i], ea[i]);
  }
  *(float4*)(ealpha + (size_t)e * HEADS) = make_float4(ea[0], ea[1], ea[2], ea[3]);
}

__global__ void invert_denom(float* __restrict__ denom) {
  int t = blockIdx.x * blockDim.x + threadIdx.x;
  if (t >= N_NODES * HEADS) return;
  denom[t] = 1.0f / denom[t];                   // >= 1 thanks to self loop
}

// one wave per edge; lane handles a float4 channel chunk (heads are 4-aligned,
// so a 4-aligned chunk never straddles a head boundary)
__global__ void edge_aggregate(const int* __restrict__ src, const int* __restrict__ dst,
                               const float* __restrict__ h,
                               const float* __restrict__ ealpha,
                               const float* __restrict__ invden,
                               float* __restrict__ acc, int Ch) {
  const int wave = threadIdx.x >> 5;
  const int lane = threadIdx.x & 31;
  const int e = blockIdx.x * (blockDim.x >> 5) + wave;
  if (e >= ETOT) return;
  int s, d;
  edge_endpoints(e, src, dst, s, d);
  const int Wd = HEADS * Ch;

  float4 ea  = *(const float4*)(ealpha + (size_t)e * HEADS);
  float4 idn = *(const float4*)(invden + (size_t)d * HEADS);
  float w[4] = {ea.x * idn.x, ea.y * idn.y, ea.z * idn.z, ea.w * idn.w};

  const float* hrow = h   + (size_t)s * Wd;
  float*       arow = acc + (size_t)d * Wd;
  for (int c0 = lane * 4; c0 < Wd; c0 += 128) {
    float4 hv = *(const float4*)(hrow + c0);
    float ww = w[c0 / Ch];
    atomicAdd(arow + c0 + 0, hv.x * ww);
    atomicAdd(arow + c0 + 1, hv.y * ww);
    atomicAdd(arow + c0 + 2, hv.z * ww);
    atomicAdd(arow + c0 + 3, hv.w * ww);
  }
}

__global__ void epilogue_relu(const float* __restrict__ acc,
                              const float* __restrict__ bias,
                              float* __restrict__ out) {
  int t = blockIdx.x * blockDim.x + threadIdx.x;
  if (t >= N_NODES * (W_HID / 4)) return;
  float4 v = *(const float4*)(acc + (size_t)t * 4);
  float4 b = *(const float4*)(bias + (t % (W_HID / 4)) * 4);
  float4 r;
  r.x = fmaxf(v.x + b.x, 0.f);
  r.y = fmaxf(v.y + b.y, 0.f);
  r.z = fmaxf(v.z + b.z, 0.f);
  r.w = fmaxf(v.w + b.w, 0.f);
  *(float4*)(out + (size_t)t * 4) = r;
}

__global__ void finalize_mean(const float* __restrict__ acc,
                              const float* __restrict__ bias,
                              float* __restrict__ out) {
  int t = blockIdx.x * blockDim.x + threadIdx.x;
  if (t >= N_NODES * (NCLS / 4)) return;
  int n = t / (NCLS / 4), q = t % (NCLS / 4);
  int c0 = q * 4;
  float4 s = make_float4(0.f, 0.f, 0.f, 0.f);
#pragma unroll
  for (int hd = 0; hd < HEADS; ++hd) {
    float4 v = *(const float4*)(acc + (size_t)n * W_OUT + hd * NCLS + c0);
    s.x += v.x; s.y += v.y; s.z += v.z; s.w += v.w;
  }
  float4 b = *(const float4*)(bias + c0);
  float4 r;
  r.x = s.x * (1.0f / HEADS) + b.x;
  r.y = s.y * (1.0f / HEADS) + b.y;
  r.z = s.z * (1.0f / HEADS) + b.z;
  r.w = s.w * (1.0f / HEADS) + b.w;
  *(float4*)(out + (size_t)n * NCLS + c0) = r;
}

// ---------------------------------------------------------------------------
extern "C" void kernel_launch(void* const* d_in, const int* in_sizes, int n_in,
                              void* d_out, int out_size, void* d_ws, size_t ws_size,
                              hipStream_t stream) {
  (void)in_sizes; (void)n_in; (void)out_size; (void)ws_size;

  const float* x  = (const float*)d_in[0];
  const int* ei   = (const int*)d_in[1];
  const float* Wm[4]  = {(const float*)d_in[2],  (const float*)d_in[6],
                         (const float*)d_in[10], (const float*)d_in[14]};
  const float* As[4]  = {(const float*)d_in[3],  (const float*)d_in[7],
                         (const float*)d_in[11], (const float*)d_in[15]};
  const float* Ad[4]  = {(const float*)d_in[4],  (const float*)d_in[8],
                         (const float*)d_in[12], (const float*)d_in[16]};
  const float* Bi[4]  = {(const float*)d_in[5],  (const float*)d_in[9],
                         (const float*)d_in[13], (const float*)d_in[17]};
  const int* src = ei;                 // edge_index row 0
  const int* dst = ei + E_EDGES;       // edge_index row 1

  // carve workspace (256-B aligned chunks)
  char* p = (char*)d_ws;
  auto carve = [&](size_t bytes) {
    void* r = (void*)p;
    p += (bytes + 255) & ~(size_t)255;
    return r;
  };
  float*    xbuf   = (float*)carve(sizeof(float) * (size_t)N_NODES * W_HID);
  float*    hbuf   = (float*)carve(sizeof(float) * (size_t)N_NODES * W_OUT);
  float*    accbuf = (float*)carve(sizeof(float) * (size_t)N_NODES * W_OUT);
  float*    es     = (float*)carve(sizeof(float) * (size_t)N_NODES * HEADS);
  float*    ed     = (float*)carve(sizeof(float) * (size_t)N_NODES * HEADS);
  unsigned* mkey   = (unsigned*)carve(sizeof(unsigned) * (size_t)N_NODES * HEADS);
  float*    denom  = (float*)carve(sizeof(float) * (size_t)N_NODES * HEADS);
  float*    ealpha = (float*)carve(sizeof(float) * (size_t)ETOT * HEADS);

  const int nh_blocks = (N_NODES * HEADS + 255) / 256;
  const int e_blocks  = (ETOT + 255) / 256;
  const int agg_blocks = (ETOT + 7) / 8;       // 8 waves (edges) per 256-thr block

  for (int L = 0; L < 4; ++L) {
    const int Ch = (L == 3) ? NCLS : CHID;
    const int Wd = HEADS * Ch;                 // 128 or 160
    const float* xin = (L == 0) ? x : xbuf;

    // 1) h = xin @ W   (WMMA f32)
    gemm_wmma_f32<<<N_NODES / 16, 32 * (Wd / 16), 0, stream>>>(xin, Wm[L], hbuf, Wd);

    // 2) reset per-layer accumulators
    hipMemsetAsync(mkey,  0, sizeof(unsigned) * (size_t)N_NODES * HEADS, stream);
    hipMemsetAsync(denom, 0, sizeof(float)    * (size_t)N_NODES * HEADS, stream);
    hipMemsetAsync(accbuf, 0, sizeof(float)   * (size_t)N_NODES * Wd,    stream);

    // 3) attention scalars
    att_scalars<<<nh_blocks, 256, 0, stream>>>(hbuf, As[L], Ad[L], es, ed, Ch);

    // 4) segment softmax (max -> exp/sum -> reciprocal)
    edge_max<<<e_blocks, 256, 0, stream>>>(src, dst, es, ed, mkey);
    edge_expsum<<<e_blocks, 256, 0, stream>>>(src, dst, es, ed, mkey, ealpha, denom);
    invert_denom<<<nh_blocks, 256, 0, stream>>>(denom);

    // 5) weighted scatter-add aggregation (dominant traffic)
    edge_aggregate<<<agg_blocks, 256, 0, stream>>>(src, dst, hbuf, ealpha, denom,
                                                   accbuf, Ch);

    // 6) epilogue
    if (L < 3) {
      epilogue_relu<<<(N_NODES * (W_HID / 4) + 255) / 256, 256, 0, stream>>>(
          accbuf, Bi[L], xbuf);
    } else {
      finalize_mean<<<(N_NODES * (NCLS / 4) + 255) / 256, 256, 0, stream>>>(
          accbuf, Bi[L], (float*)d_out);
    }
  }
}